// TemporalResonanceLayer_66855460930128
// MI455X (gfx1250) — compile-verified
//
#include <hip/hip_runtime.h>
#include <math.h>

// ---- problem constants (from reference) ----
#define B_SZ   4
#define N_SEQ  4096
#define D_DIM  1024
#define K_DIM  256
#define P_DIM  (6*K_DIM)     // 1536
#define K2_DIM (2*K_DIM)     // 512
#define NC     64            // scan chunks along N
#define CL     (N_SEQ/NC)    // 64 steps per chunk
#define GATE_BIAS 0.6190392084062235f   // log(0.65/0.35)
#define AMP_MAX   3.0f

typedef __attribute__((ext_vector_type(2))) float v2f;
typedef __attribute__((ext_vector_type(8))) float v8f;

// 16-byte int vector (gcc-style, matches the async-LDS builtin prototype)
typedef int v4i_ __attribute__((vector_size(16)));
typedef __attribute__((address_space(1))) v4i_ g_v4i;   // global
typedef __attribute__((address_space(3))) v4i_ l_v4i;   // LDS

#if defined(__HIP_DEVICE_COMPILE__) && __has_builtin(__builtin_amdgcn_global_load_async_to_lds_b128)
#define HAVE_ASYNC_LDS 1
#else
#define HAVE_ASYNC_LDS 0
#endif

// 16-byte global -> LDS copy. Async (ASYNCcnt-tracked, no VGPR roundtrip)
// when the gfx1250 builtin is available; synchronous fallback otherwise.
__device__ __forceinline__ void cp16_g2l(const float* g, float* l) {
#if HAVE_ASYNC_LDS
    __builtin_amdgcn_global_load_async_to_lds_b128(
        (g_v4i*)(uintptr_t)g,
        (l_v4i*)(uintptr_t)l,   // flat->LDS = low 32 bits
        0, 0);
#else
    *(float4*)l = *(const float4*)g;
#endif
}

template <int N>
__device__ __forceinline__ void wait_async() {
#if HAVE_ASYNC_LDS
    __builtin_amdgcn_s_wait_asynccnt(N);
#endif
}

__device__ __forceinline__ float sigmoidf_(float x) { return 1.0f / (1.0f + expf(-x)); }

// =====================================================================
// Kernel 1: raw = x @ W_proj^T + b_proj, fused activation epilogue.
// Block = 128 threads (4 waves). Each block: 64 rows (M) x 16 channels,
// computing ALL 6 heads for those channels so the epilogue can produce
// alpha / drive_r / drive_i / cos / sin / g / beta directly.
// fp32 WMMA 16x16x4; K-chunks of 32 streamed into double-buffered LDS
// via async global->LDS copies.
// =====================================================================
__global__ __launch_bounds__(128) void k_gemm1_act(
    const float* __restrict__ x, const float* __restrict__ Wp, const float* __restrict__ bp,
    float* __restrict__ wAlpha, float* __restrict__ wDr, float* __restrict__ wDi,
    float* __restrict__ wCos, float* __restrict__ wSin,
    float* __restrict__ wG, float* __restrict__ wBeta)
{
    __shared__ float As[2][64][36];   // 64 M-rows x 32 K, stride 36 (16B-aligned, conflict-free)
    __shared__ float Bs[2][96][36];   // 6 heads x 16 channels x 32 K

    const int tid  = threadIdx.x;
    const int w    = tid >> 5;        // wave 0..3 -> M sub-tile
    const int lane = tid & 31;
    const int lm   = lane & 15;
    const int lh   = lane >> 4;
    const int m0   = blockIdx.x * 64;        // global M (= b*N + n) tile base
    const int kch0 = blockIdx.y * 16;        // channel tile base within K_DIM

    v8f acc[6];
#pragma unroll
    for (int h = 0; h < 6; ++h) acc[h] = (v8f)0.0f;

    const int arow = tid >> 3;           // 0..15
    const int acg  = (tid & 7) * 4;      // column group * 4

    // stage one K-chunk (10 x b128 per thread: 4 A rows + 6 B rows)
    auto issue = [&](int buf, int kc) {
#pragma unroll
        for (int rr = 0; rr < 4; ++rr) {
            const int r = arow + rr * 16;
            cp16_g2l(&x[(size_t)(m0 + r) * D_DIM + kc + acg], &As[buf][r][acg]);
        }
#pragma unroll
        for (int rr = 0; rr < 6; ++rr) {
            const int r = arow + rr * 16;                  // head = rr, channel = arow
            cp16_g2l(&Wp[(size_t)(rr * K_DIM + kch0 + arow) * D_DIM + kc + acg],
                     &Bs[buf][r][acg]);
        }
    };

    constexpr int NCHUNK = D_DIM / 32;
    issue(0, 0);
    for (int ic = 0; ic < NCHUNK; ++ic) {
        const int cur = ic & 1;
        if (ic + 1 < NCHUNK) {
            issue(cur ^ 1, (ic + 1) * 32);   // prefetch next chunk into other buffer
            wait_async<10>();                // current chunk's 10 copies complete
        } else {
            wait_async<0>();
        }
        __syncthreads();

#pragma unroll
        for (int kk = 0; kk < 32; kk += 4) {
            const v2f af = *(const v2f*)&As[cur][w * 16 + lm][kk + 2 * lh];
#pragma unroll
            for (int h = 0; h < 6; ++h) {
                const v2f bf = *(const v2f*)&Bs[cur][h * 16 + lm][kk + 2 * lh];
                acc[h] = __builtin_amdgcn_wmma_f32_16x16x4_f32(
                    false, af, false, bf, (short)0, acc[h], false, false);
            }
        }
        __syncthreads();   // all waves done reading buf[cur] before it is refilled
    }

    // ---- epilogue: bias + activations + oscillator drive ----
    const int k = kch0 + lm;
    float bias[6];
#pragma unroll
    for (int h = 0; h < 6; ++h) bias[h] = bp[h * K_DIM + k];

#pragma unroll
    for (int j = 0; j < 8; ++j) {
        const int gm = m0 + w * 16 + j + 8 * lh;   // global row = b*N + n
        const int n  = gm & (N_SEQ - 1);
        const size_t idx = (size_t)gm * K_DIM + k;

        const float rA  = acc[0][j] + bias[0];
        const float rO  = acc[1][j] + bias[1];
        const float phi = acc[2][j] + bias[2];
        const float rAl = acc[3][j] + bias[3];
        const float rG  = acc[4][j] + bias[4];
        const float rBe = acc[5][j] + bias[5];

        const float A     = AMP_MAX * sigmoidf_(rA);
        const float omega = (rO > 20.0f) ? rO : log1pf(expf(rO));   // softplus
        const float alpha = sigmoidf_(rAl + GATE_BIAS);
        const float g     = sigmoidf_(rG);
        const float beta  = sigmoidf_(rBe);

        const float pos = log1pf((float)n);
        const float ang = omega * pos + phi;
        float s, c;
        sincosf(ang, &s, &c);

        const float oma = 1.0f - alpha;
        wAlpha[idx] = alpha;
        wDr[idx]    = oma * A * c;
        wDi[idx]    = oma * A * s;
        wCos[idx]   = c;
        wSin[idx]   = s;
        wG[idx]     = g;
        wBeta[idx]  = beta;
    }
}

// =====================================================================
// Kernel 2: per-chunk scan aggregates. One thread per (b, chunk, k):
// A = prod(alpha), (Rr,Ri) = chunk-local scan result with zero carry.
// =====================================================================
__global__ __launch_bounds__(256) void k_scan_agg(
    const float* __restrict__ wAlpha, const float* __restrict__ wDr, const float* __restrict__ wDi,
    float* __restrict__ aggA, float* __restrict__ aggR, float* __restrict__ aggI)
{
    const int k     = threadIdx.x;
    const int chunk = blockIdx.x & (NC - 1);
    const int b     = blockIdx.x >> 6;            // NC == 64
    const size_t base = ((size_t)b * N_SEQ + (size_t)chunk * CL) * K_DIM + k;

    float A = 1.0f, Rr = 0.0f, Ri = 0.0f;
    for (int t = 0; t < CL; ++t) {
        const size_t i = base + (size_t)t * K_DIM;
        const float a = wAlpha[i];
        Rr = fmaf(a, Rr, wDr[i]);
        Ri = fmaf(a, Ri, wDi[i]);
        A *= a;
    }
    const size_t o = (size_t)(b * NC + chunk) * K_DIM + k;
    aggA[o] = A; aggR[o] = Rr; aggI[o] = Ri;
}

// =====================================================================
// Kernel 3: exclusive scan of chunk aggregates along the chunk axis.
// One thread per (b, k): 64 sequential combines.
// =====================================================================
__global__ __launch_bounds__(256) void k_scan_carry(
    const float* __restrict__ aggA, const float* __restrict__ aggR, const float* __restrict__ aggI,
    float* __restrict__ carR, float* __restrict__ carI)
{
    const int k = threadIdx.x;
    const int b = blockIdx.x;
    float cr = 0.0f, ci = 0.0f;
    for (int c = 0; c < NC; ++c) {
        const size_t o = (size_t)(b * NC + c) * K_DIM + k;
        carR[o] = cr; carI[o] = ci;
        const float A = aggA[o];
        cr = fmaf(A, cr, aggR[o]);
        ci = fmaf(A, ci, aggI[o]);
    }
}

// =====================================================================
// Kernel 4: re-scan each chunk seeded with its carry, fused with the
// erase / normalize / demodulate / gate post-processing -> rho (B,N,2K).
// =====================================================================
__global__ __launch_bounds__(256) void k_scan_apply(
    const float* __restrict__ wAlpha, const float* __restrict__ wDr, const float* __restrict__ wDi,
    const float* __restrict__ wCos, const float* __restrict__ wSin,
    const float* __restrict__ wG, const float* __restrict__ wBeta,
    const float* __restrict__ carR, const float* __restrict__ carI,
    float* __restrict__ rho)
{
    const int k     = threadIdx.x;
    const int chunk = blockIdx.x & (NC - 1);
    const int b     = blockIdx.x >> 6;
    const size_t o  = (size_t)(b * NC + chunk) * K_DIM + k;

    float rr = carR[o], ri = carI[o];
    const int n0 = chunk * CL;

    for (int t = 0; t < CL; ++t) {
        const int n = n0 + t;
        const size_t i = ((size_t)b * N_SEQ + n) * K_DIM + k;
        const float a = wAlpha[i];
        rr = fmaf(a, rr, wDr[i]);
        ri = fmaf(a, ri, wDi[i]);

        const float c = wCos[i], s = wSin[i];
        const float beta = wBeta[i], g = wG[i];

        const float readout = rr * c + ri * s;
        float r2 = rr - beta * readout * c;
        float i2 = ri - beta * readout * s;

        const float mod = sqrtf(r2 * r2 + i2 * i2 + 1e-8f);
        const float sc  = fmaxf(mod, 1.0f);
        r2 /= sc; i2 /= sc;

        const float re =  r2 * c + i2 * s;
        const float im = -r2 * s + i2 * c;

        const size_t ro = ((size_t)b * N_SEQ + n) * K2_DIM + k;
        rho[ro]         = g * re;
        rho[ro + K_DIM] = g * im;
    }
}

// =====================================================================
// Kernel 5: out = res_scale * (rho @ W_res^T). M=16384, N(d)=1024, K=512.
// Block = 256 threads (8 waves): 4 M-tiles x 2 D-tiles of 16x16 each.
// Async double-buffered LDS staging (3 x b128 per thread per chunk).
// =====================================================================
__global__ __launch_bounds__(256) void k_gemm2(
    const float* __restrict__ rho, const float* __restrict__ Wr,
    const float* __restrict__ res_scale, float* __restrict__ out)
{
    __shared__ float As[2][64][36];
    __shared__ float Bs[2][32][36];

    const int tid  = threadIdx.x;
    const int w    = tid >> 5;
    const int lane = tid & 31;
    const int lm   = lane & 15;
    const int lh   = lane >> 4;
    const int mt   = w & 3;
    const int dt   = w >> 2;
    const int m0   = blockIdx.x * 64;
    const int d0   = blockIdx.y * 32;

    v8f acc = (v8f)0.0f;

    const int arow = tid >> 3;          // 0..31
    const int acg  = (tid & 7) * 4;

    auto issue = [&](int buf, int kc) {
#pragma unroll
        for (int rr = 0; rr < 2; ++rr) {
            const int r = arow + rr * 32;
            cp16_g2l(&rho[(size_t)(m0 + r) * K2_DIM + kc + acg], &As[buf][r][acg]);
        }
        cp16_g2l(&Wr[(size_t)(d0 + arow) * K2_DIM + kc + acg], &Bs[buf][arow][acg]);
    };

    constexpr int NCHUNK = K2_DIM / 32;
    issue(0, 0);
    for (int ic = 0; ic < NCHUNK; ++ic) {
        const int cur = ic & 1;
        if (ic + 1 < NCHUNK) {
            issue(cur ^ 1, (ic + 1) * 32);
            wait_async<3>();
        } else {
            wait_async<0>();
        }
        __syncthreads();

#pragma unroll
        for (int kk = 0; kk < 32; kk += 4) {
            const v2f af = *(const v2f*)&As[cur][mt * 16 + lm][kk + 2 * lh];
            const v2f bf = *(const v2f*)&Bs[cur][dt * 16 + lm][kk + 2 * lh];
            acc = __builtin_amdgcn_wmma_f32_16x16x4_f32(
                false, af, false, bf, (short)0, acc, false, false);
        }
        __syncthreads();
    }

    const float sc  = res_scale[0];
    const int dcol  = d0 + dt * 16 + lm;
#pragma unroll
    for (int j = 0; j < 8; ++j) {
        const int gm = m0 + mt * 16 + j + 8 * lh;
        out[(size_t)gm * D_DIM + dcol] = sc * acc[j];
    }
}

// =====================================================================
extern "C" void kernel_launch(void* const* d_in, const int* in_sizes, int n_in,
                              void* d_out, int out_size, void* d_ws, size_t ws_size,
                              hipStream_t stream) {
    const float* x  = (const float*)d_in[0];   // (B,N,D)
    const float* Wp = (const float*)d_in[1];   // (6K,D)
    const float* bp = (const float*)d_in[2];   // (6K,)
    const float* Wr = (const float*)d_in[3];   // (D,2K)
    const float* rs = (const float*)d_in[4];   // scalar
    float* out = (float*)d_out;                // (B,N,D)

    const size_t SZ1  = (size_t)B_SZ * N_SEQ * K_DIM;   // 4,194,304
    const size_t SZA  = (size_t)B_SZ * NC * K_DIM;      // 65,536
    float* p = (float*)d_ws;
    float* wAlpha = p; p += SZ1;
    float* wDr    = p; p += SZ1;
    float* wDi    = p; p += SZ1;
    float* wCos   = p; p += SZ1;
    float* wSin   = p; p += SZ1;
    float* wG     = p; p += SZ1;
    float* wBeta  = p; p += SZ1;
    float* aggA   = p; p += SZA;
    float* aggR   = p; p += SZA;
    float* aggI   = p; p += SZA;
    float* carR   = p; p += SZA;
    float* carI   = p; p += SZA;
    float* rho    = p; p += (size_t)B_SZ * N_SEQ * K2_DIM;

    // 1) projection GEMM + activations (fp32 WMMA, async LDS staging)
    dim3 g1((B_SZ * N_SEQ) / 64, K_DIM / 16);
    k_gemm1_act<<<g1, 128, 0, stream>>>(x, Wp, bp, wAlpha, wDr, wDi, wCos, wSin, wG, wBeta);

    // 2) chunked associative scan
    k_scan_agg  <<<B_SZ * NC, 256, 0, stream>>>(wAlpha, wDr, wDi, aggA, aggR, aggI);
    k_scan_carry<<<B_SZ,      256, 0, stream>>>(aggA, aggR, aggI, carR, carI);
    k_scan_apply<<<B_SZ * NC, 256, 0, stream>>>(wAlpha, wDr, wDi, wCos, wSin, wG, wBeta,
                                                carR, carI, rho);

    // 3) residual GEMM (fp32 WMMA) with res_scale epilogue
    dim3 g2((B_SZ * N_SEQ) / 64, D_DIM / 32);
    k_gemm2<<<g2, 256, 0, stream>>>(rho, Wr, rs, out);
}